// RQBottleneck_29222957482313
// MI455X (gfx1250) — compile-verified
//
#include <hip/hip_runtime.h>

typedef float v2f __attribute__((ext_vector_type(2)));
typedef float v8f __attribute__((ext_vector_type(8)));

#define D        256
#define KCODES   4096
#define LEVELS   4
#define NROWS    32768          /* 8*4096 */
#define WAVES_PER_WG 8
#define ROWS_PER_WG  (WAVES_PER_WG * 16)   /* 128 */
#define TILE_N   16
#define NTILES   (KCODES / TILE_N)         /* 256 */
#define LDS_STRIDE 260          /* floats; 260 % 64 == 4 -> conflict-free b64 reads */

// ---------------------------------------------------------------------------
// Fused distance + argmin over one level: for each row n of `residual`,
// codes[n] = argmin_k ( |c_k|^2 - 2 * r_n . c_k )   (|r|^2 dropped: row-const)
// Double-buffered LDS staging via async global->LDS (ASYNCcnt) overlapped
// with the f32 WMMA stream.
// ---------------------------------------------------------------------------
__global__ __launch_bounds__(256) void rvq_dist_argmin(
    const float* __restrict__ residual,   // [NROWS, D]
    const float* __restrict__ cb,         // [KCODES, D] (this level)
    const float* __restrict__ norms,      // [KCODES]    (this level)
    int*         __restrict__ codes,      // [NROWS]
    float*       __restrict__ codes_out,  // [NROWS, LEVELS] float codes in d_out
    int level)
{
  __shared__ float bt[2][TILE_N * LDS_STRIDE];

  const int tid  = threadIdx.x;
  const int wave = tid >> 5;
  const int lane = tid & 31;
  const int half = lane >> 4;   // 0: K={0,1}, 1: K={2,3} within a 4-wide k-chunk
  const int n16  = lane & 15;

  const int row_base = blockIdx.x * ROWS_PER_WG + wave * 16;
  const int arow     = row_base + n16;     // A-matrix row held by this lane

  // A = -2 * residual rows, resident in registers in WMMA 16x4 f32 layout.
  v2f a[64];
  {
    const float* rp = residual + (size_t)arow * D + 2 * half;
    #pragma unroll
    for (int kc = 0; kc < 64; ++kc) {
      a[kc].x = -2.0f * rp[4 * kc + 0];
      a[kc].y = -2.0f * rp[4 * kc + 1];
    }
  }

  // Async staging parameters: 32 B128s per tile per WG => 4 per wave (ASYNCcnt=4).
  // Flat float4 index g = i*256 + tid ; code = g>>6 ; col4 = g&63 (contiguous 512B
  // per instruction across the 32 lanes of a wave).
  unsigned      lds_off[4];
  const char*   gsrc[4];
  #pragma unroll
  for (int i = 0; i < 4; ++i) {
    int g    = i * 256 + tid;
    int code = g >> 6;                       // 64 float4 per code row
    int col4 = g & 63;
    lds_off[i] = (unsigned)((code * LDS_STRIDE + col4 * 4) * 4);   // bytes
    gsrc[i]    = (const char*)cb + (size_t)g * 16;                 // tile 0 base
  }
  const unsigned lds_base[2] = { (unsigned)(size_t)(&bt[0][0]),
                                 (unsigned)(size_t)(&bt[1][0]) };

  // Prologue: kick off tile 0 into buffer 0.
  #pragma unroll
  for (int i = 0; i < 4; ++i) {
    unsigned    lda = lds_base[0] + lds_off[i];
    const void* ga  = gsrc[i];
    asm volatile("global_load_async_to_lds_b128 %0, %1, off"
                 :: "v"(lda), "v"(ga) : "memory");
  }

  float minval[8];
  int   minidx[8];
  #pragma unroll
  for (int m = 0; m < 8; ++m) { minval[m] = 3.4e38f; minidx[m] = 0; }

  for (int t = 0; t < NTILES; ++t) {
    // All waves finished reading buffer (t+1)&1 (during iteration t-1).
    __syncthreads();

    if (t + 1 < NTILES) {
      // Issue tile t+1 into the other buffer, then wait until only those 4
      // remain outstanding (in-order completion => tile t's writes landed).
      const size_t adv = (size_t)(t + 1) * TILE_N * D * sizeof(float);
      #pragma unroll
      for (int i = 0; i < 4; ++i) {
        unsigned    lda = lds_base[(t + 1) & 1] + lds_off[i];
        const void* ga  = gsrc[i] + adv;
        asm volatile("global_load_async_to_lds_b128 %0, %1, off"
                     :: "v"(lda), "v"(ga) : "memory");
      }
      asm volatile("s_wait_asynccnt 0x4" ::: "memory");
    } else {
      asm volatile("s_wait_asynccnt 0x0" ::: "memory");
    }
    __syncthreads();   // tile t fully visible to every wave

    const float nc = norms[t * TILE_N + n16];   // |c|^2 for this lane's column

    // 16x16 tile of -2*r.c via 64 chained f32 WMMAs (two accumulators for ILP)
    v8f c0 = {0.f,0.f,0.f,0.f,0.f,0.f,0.f,0.f};
    v8f c1 = {0.f,0.f,0.f,0.f,0.f,0.f,0.f,0.f};
    const float* bp = &bt[t & 1][n16 * LDS_STRIDE + 2 * half];
    #pragma unroll
    for (int kc = 0; kc < 64; kc += 2) {
      v2f b0; b0.x = bp[4 * kc + 0]; b0.y = bp[4 * kc + 1];
      c0 = __builtin_amdgcn_wmma_f32_16x16x4_f32(false, a[kc],     false, b0,
                                                 (short)0, c0, false, false);
      v2f b1; b1.x = bp[4 * kc + 4]; b1.y = bp[4 * kc + 5];
      c1 = __builtin_amdgcn_wmma_f32_16x16x4_f32(false, a[kc + 1], false, b1,
                                                 (short)0, c1, false, false);
    }

    #pragma unroll
    for (int m = 0; m < 8; ++m) {
      float d = c0[m] + c1[m] + nc;
      if (d < minval[m]) { minval[m] = d; minidx[m] = t * TILE_N + n16; }
    }
  }

  // Cross-lane argmin over the 16 lanes (N dimension) holding the same row.
  #pragma unroll
  for (int m = 0; m < 8; ++m) {
    float v   = minval[m];
    int   idx = minidx[m];
    #pragma unroll
    for (int off = 8; off >= 1; off >>= 1) {
      float v2 = __shfl_xor(v,   off, 16);
      int   i2 = __shfl_xor(idx, off, 16);
      if (v2 < v || (v2 == v && i2 < idx)) { v = v2; idx = i2; }
    }
    if (n16 == 0) {
      int row = row_base + m + 8 * half;   // C vgpr m: lanes<16 -> M=m, else M=m+8
      codes[row] = idx;
      codes_out[(size_t)row * LEVELS + level] = (float)idx;
    }
  }
}

// ---------------------------------------------------------------------------
// Gather q = cb[codes[n]]; residual -= q; z_q (+)= q; commit += sum(residual^2)
// One wave per row; 8 rows per block.
// ---------------------------------------------------------------------------
__global__ __launch_bounds__(256) void rvq_update(
    const float* __restrict__ cb,        // [KCODES, D]
    const int*   __restrict__ codes,     // [NROWS]
    float*       __restrict__ residual,  // [NROWS, D]
    float*       __restrict__ zq,        // d_out z_q region [NROWS, D]
    float*       __restrict__ commit,    // d_out scalar
    int level)
{
  __shared__ float red[8];
  const int tid  = threadIdx.x;
  const int wave = tid >> 5;
  const int lane = tid & 31;
  const int row  = blockIdx.x * 8 + wave;
  const int idx  = codes[row];

  const float4* q4 = (const float4*)(cb + (size_t)idx * D);
  float4*       r4 = (float4*)(residual + (size_t)row * D);
  float4*       z4 = (float4*)(zq + (size_t)row * D);

  float s = 0.0f;
  #pragma unroll
  for (int i = 0; i < 2; ++i) {
    int    j = lane + i * 32;            // 64 float4 per row
    float4 q = q4[j];
    float4 r = r4[j];
    r.x -= q.x; r.y -= q.y; r.z -= q.z; r.w -= q.w;
    r4[j] = r;
    float4 z;
    if (level == 0) { z = q; }
    else { z = z4[j]; z.x += q.x; z.y += q.y; z.z += q.z; z.w += q.w; }
    z4[j] = z;
    s += r.x * r.x + r.y * r.y + r.z * r.z + r.w * r.w;
  }
  #pragma unroll
  for (int off = 16; off >= 1; off >>= 1) s += __shfl_xor(s, off, 32);
  if (lane == 0) red[wave] = s;
  __syncthreads();
  if (tid == 0) {
    float tsum = 0.0f;
    #pragma unroll
    for (int w = 0; w < 8; ++w) tsum += red[w];
    atomicAdd(commit, tsum * (1.0f / ((float)LEVELS * (float)NROWS * (float)D)));
  }
}

// ---------------------------------------------------------------------------
// norms[l*K + k] = |cb[l][k]|^2   (one block per code row)
// ---------------------------------------------------------------------------
__global__ __launch_bounds__(256) void rvq_norms(
    const float* __restrict__ cbs, float* __restrict__ norms)
{
  __shared__ float red[8];
  const int tid = threadIdx.x;
  const float v = cbs[(size_t)blockIdx.x * D + tid];
  float s = v * v;
  #pragma unroll
  for (int off = 16; off >= 1; off >>= 1) s += __shfl_xor(s, off, 32);
  if ((tid & 31) == 0) red[tid >> 5] = s;
  __syncthreads();
  if (tid == 0) {
    float t = 0.0f;
    #pragma unroll
    for (int w = 0; w < 8; ++w) t += red[w];
    norms[blockIdx.x] = t;
  }
}

// residual = x ; commit = 0
__global__ void rvq_init(const float4* __restrict__ x, float4* __restrict__ r,
                         float* __restrict__ commit, int n4)
{
  int i = blockIdx.x * blockDim.x + threadIdx.x;
  if (i < n4) r[i] = x[i];
  if (i == 0) *commit = 0.0f;
}

extern "C" void kernel_launch(void* const* d_in, const int* in_sizes, int n_in,
                              void* d_out, int out_size, void* d_ws, size_t ws_size,
                              hipStream_t stream) {
  const float* x   = (const float*)d_in[0];   // [8,4096,256]
  const float* cbs = (const float*)d_in[1];   // [4,4096,256]

  float* zq        = (float*)d_out;                   // NROWS*D floats
  float* commit    = zq + (size_t)NROWS * D;          // 1 float
  float* codes_out = commit + 1;                      // NROWS*LEVELS floats

  char* ws = (char*)d_ws;
  float* residual = (float*)ws;                                       // 32 MB
  int*   codes    = (int*)(ws + (size_t)NROWS * D * sizeof(float));   // 128 KB
  float* norms    = (float*)(ws + (size_t)NROWS * D * sizeof(float)
                                + (size_t)NROWS * sizeof(int));       // 64 KB

  const int n4 = NROWS * D / 4;
  rvq_init<<<(n4 + 255) / 256, 256, 0, stream>>>((const float4*)x,
                                                 (float4*)residual, commit, n4);
  rvq_norms<<<LEVELS * KCODES, 256, 0, stream>>>(cbs, norms);

  for (int l = 0; l < LEVELS; ++l) {
    const float* cb = cbs + (size_t)l * KCODES * D;
    rvq_dist_argmin<<<NROWS / ROWS_PER_WG, 256, 0, stream>>>(
        residual, cb, norms + (size_t)l * KCODES, codes, codes_out, l);
    rvq_update<<<NROWS / 8, 256, 0, stream>>>(
        cb, codes, residual, zq, commit, l);
  }
}